// TextDecoder_42039139893533
// MI455X (gfx1250) — compile-verified
//
#include <hip/hip_runtime.h>
#include <hip/hip_bf16.h>

typedef __attribute__((ext_vector_type(2))) float v2f;
typedef __attribute__((ext_vector_type(8))) float v8f;

// Workspace layout (float offsets)
#define WS_T1   0     // sum(W1)
#define WS_BS1  1     // sum(b1)
#define WS_T2   2     // sum(W2)
#define WS_BS2  3     // sum(b2)
#define WS_S1   16    // 512 floats: row sums of z1
#define WS_A    1024  // 512 floats: collapsed per-(b,l) scalar
#define WS_CW3  2048  // 768 floats: column sums of W3

// ---------- total sums of W1, b1, W2, b2 (deterministic tree) ----------
__global__ void total_sums_kernel(const float* __restrict__ W1,
                                  const float* __restrict__ b1,
                                  const float* __restrict__ W2,
                                  const float* __restrict__ b2,
                                  float* __restrict__ ws) {
    __shared__ float sm[256];
    const float* src; int n; int slot;
    switch (blockIdx.x) {
        case 0:  src = W1; n = 256 * 512; slot = WS_T1;  break;
        case 1:  src = b1; n = 512;       slot = WS_BS1; break;
        case 2:  src = W2; n = 512 * 768; slot = WS_T2;  break;
        default: src = b2; n = 768;       slot = WS_BS2; break;
    }
    float acc = 0.0f;
    for (int i = threadIdx.x; i < n; i += 256) acc += src[i];  // fixed order
    sm[threadIdx.x] = acc;
    __syncthreads();
    for (int s = 128; s > 0; s >>= 1) {
        if (threadIdx.x < s) sm[threadIdx.x] += sm[threadIdx.x + s];
        __syncthreads();
    }
    if (threadIdx.x == 0) ws[slot] = sm[0];
}

// ---------- S1[b,l]: row sums of z1 (512 rows x 256) ----------
__global__ void row_sums_kernel(const float* __restrict__ z1,
                                float* __restrict__ ws) {
    __shared__ float sm[256];
    const int row = blockIdx.x;                  // 0..511
    sm[threadIdx.x] = z1[row * 256 + threadIdx.x];
    __syncthreads();
    for (int s = 128; s > 0; s >>= 1) {
        if (threadIdx.x < s) sm[threadIdx.x] += sm[threadIdx.x + s];
        __syncthreads();
    }
    if (threadIdx.x == 0) ws[WS_S1 + row] = sm[0];
}

// ---------- CW3[g]: column sums of W3 (768x768) via V_WMMA_F32_16X16X4_F32 ----
// ones(16x4) @ W3_tile(4x16) accumulated over K=768 -> every D row = column sums.
// All-ones A makes the result invariant to B's K-half lane assignment.
__global__ void colsum_wmma_kernel(const float* __restrict__ W3,
                                   float* __restrict__ ws) {
    const int lane = threadIdx.x;        // one wave32 per block
    const int g0   = blockIdx.x * 16;    // 48 blocks x 16 columns
    const int grp  = lane >> 4;          // 0 or 1
    const int col  = lane & 15;
    const float* base = W3 + g0 + col;

    v2f a; a[0] = 1.0f; a[1] = 1.0f;     // all-ones A matrix
    v8f c0 = {};
    v8f c1 = {};
    for (int k = 0; k < 768; k += 8) {
        v2f b0, b1v;
        // chunk 0: rows k .. k+3 (each lane-half loads 2 distinct rows)
        b0[0]  = base[(k + 2 * grp    ) * 768];
        b0[1]  = base[(k + 2 * grp + 1) * 768];
        // chunk 1: rows k+4 .. k+7
        b1v[0] = base[(k + 4 + 2 * grp    ) * 768];
        b1v[1] = base[(k + 4 + 2 * grp + 1) * 768];
        c0 = __builtin_amdgcn_wmma_f32_16x16x4_f32(false, a, false, b0,
                                                   (short)0, c0, false, false);
        c1 = __builtin_amdgcn_wmma_f32_16x16x4_f32(false, a, false, b1v,
                                                   (short)0, c1, false, false);
    }
    // D layout: VGPR0, lanes 0-15 -> M=0, N=lane. Row 0 holds the column sums.
    if (lane < 16) ws[WS_CW3 + g0 + lane] = c0[0] + c1[0];
}

// ---------- A[b,l] = (65536*S1*T1 + sum(b1))*T2 + sum(b2) ----------
__global__ void finalize_A_kernel(float* __restrict__ ws) {
    const int i = threadIdx.x;           // one block of 512
    const float T1  = ws[WS_T1];
    const float Bs1 = ws[WS_BS1];
    const float T2  = ws[WS_T2];
    const float Bs2 = ws[WS_BS2];
    if (i < 512) {
        const float s1 = ws[WS_S1 + i];
        ws[WS_A + i] = fmaf(fmaf(65536.0f * s1, T1, Bs1), T2, Bs2);
    }
}

// ---------- out[b,r,g] = A[b, r//64]*CW3[g] + b3[g] : 100 MB coalesced ----------
// blockDim = 192 (float4 per 768-col row); each block writes 4 consecutive rows.
__global__ void broadcast_out_kernel(const float* __restrict__ ws,
                                     const float* __restrict__ b3,
                                     float* __restrict__ out) {
    const int t = threadIdx.x;
    const float4 cw = ((const float4*)(ws + WS_CW3))[t];
    const float4 bb = ((const float4*)b3)[t];
    float4* __restrict__ out4 = (float4*)out;
    const int row0 = blockIdx.x * 4;
#pragma unroll
    for (int j = 0; j < 4; ++j) {
        const int row = row0 + j;                 // 0 .. 32767
        const float a = ws[WS_A + (row >> 6)];
        float4 o;
        o.x = fmaf(a, cw.x, bb.x);
        o.y = fmaf(a, cw.y, bb.y);
        o.z = fmaf(a, cw.z, bb.z);
        o.w = fmaf(a, cw.w, bb.w);
        out4[row * 192 + t] = o;                  // global_store_b128
    }
}

extern "C" void kernel_launch(void* const* d_in, const int* in_sizes, int n_in,
                              void* d_out, int out_size, void* d_ws, size_t ws_size,
                              hipStream_t stream) {
    const float* z1 = (const float*)d_in[0];
    const float* W1 = (const float*)d_in[1];
    const float* b1 = (const float*)d_in[2];
    const float* W2 = (const float*)d_in[3];
    const float* b2 = (const float*)d_in[4];
    const float* W3 = (const float*)d_in[5];
    const float* b3 = (const float*)d_in[6];
    float* ws  = (float*)d_ws;
    float* out = (float*)d_out;

    total_sums_kernel<<<4, 256, 0, stream>>>(W1, b1, W2, b2, ws);
    row_sums_kernel<<<512, 256, 0, stream>>>(z1, ws);
    colsum_wmma_kernel<<<48, 32, 0, stream>>>(W3, ws);
    finalize_A_kernel<<<1, 512, 0, stream>>>(ws);
    broadcast_out_kernel<<<8192, 192, 0, stream>>>(ws, b3, out);
}